// PatchInfoGainLoss_82085414961155
// MI455X (gfx1250) — compile-verified
//
#include <hip/hip_runtime.h>
#include <hip/hip_bf16.h>

// CDNA5 / gfx1250. Wave32. Patch KDE entropy with the 64-pixel reduction done
// on the matrix pipe via V_WMMA_F32_16X16X4_F32 (full-precision f32 WMMA).

typedef float v2f __attribute__((ext_vector_type(2)));
typedef float v8f __attribute__((ext_vector_type(8)));

#define PE_EPS      1e-10f
#define PE_NEGHALF_INVBW2 (-0.08f)   /* -0.5 / (2.5*2.5) */

__global__ __launch_bounds__(256)
void PatchInfoGainLoss_82085414961155_kernel(const float* __restrict__ images,
                                             float* __restrict__ out)
{
    const int lane = threadIdx.x & 31;          // wave32 lane
    const int wave = threadIdx.x >> 5;          // 8 waves / block
    const int P    = blockIdx.x * 8 + wave;     // patch id 0..2047

    // P -> (img = n*4+sf, patch row ph, patch col pw); depth channel is C=3 of 4.
    const int  pw   = P & 15;
    const int  ph   = (P >> 4) & 15;
    const int  img  = P >> 8;                   // 0..7
    const long base = (long)(img * 4 + 3) * (128L * 128L);

    // Each lane loads pixels 2L and 2L+1 of its patch (row-major within 8x8 patch).
    float x0, x1;
    {
        const int p0  = 2 * lane;
        const int r   = p0 >> 3;
        const int c   = p0 & 7;
        const long off = (long)(ph * 8 + r) * 128 + (pw * 8 + c);
        x0 = images[base + off];
        x1 = images[base + off + 1];        // 2L,2L+1 always share a patch row
    }

    const int   h      = lane >> 4;             // lane-half: both halves cover N=0..15
    const float bin_lo = (float)(lane & 15);    // this lane's bin column within a group
    const v2f   a_ones = {1.0f, 1.0f};          // A = ones -> D rows = column sums of B

    float s[16];                                // unnormalized pdf, bin = 16*g + (lane&15)

    #pragma unroll
    for (int g = 0; g < 16; ++g) {
        const float b = (float)(g << 4) + bin_lo;
        v8f c = {0.f, 0.f, 0.f, 0.f, 0.f, 0.f, 0.f, 0.f};
        #pragma unroll
        for (int t = 0; t < 16; ++t) {
            // Broadcast pixels 4t+2h and 4t+2h+1 (held by lane 2t+h) to all lanes.
            const float xv0 = __shfl(x0, 2 * t + h, 32);
            const float xv1 = __shfl(x1, 2 * t + h, 32);
            const float d0  = xv0 - b;
            const float d1  = xv1 - b;
            v2f kb;
            kb.x = __expf(PE_NEGHALF_INVBW2 * d0 * d0);
            kb.y = __expf(PE_NEGHALF_INVBW2 * d1 * d1);
            // D = A(ones, 16x4) * B(kern, 4x16) + C : accumulates sum over 4 pixels
            // per column. Any K-row permutation of B gives the same column sum, so
            // this is robust to the exact (VGPR, lane-half) -> K mapping.
            c = __builtin_amdgcn_wmma_f32_16x16x4_f32(
                    /*neg_a=*/false, a_ones,
                    /*neg_b=*/false, kb,
                    /*c_mod=*/(short)0, c,
                    /*reuse_a=*/false, /*reuse_b=*/false);
        }
        s[g] = c[0];   // row M=0 (lanes 0-15) / M=8 (lanes 16-31): all rows identical
    }

    // Total mass over all 256 bins: per-lane sum over its 16 groups, then a
    // 16-lane xor tree (each 16-lane half holds all N=0..15 columns).
    float tot = 0.0f;
    #pragma unroll
    for (int g = 0; g < 16; ++g) tot += s[g];
    #pragma unroll
    for (int off = 1; off <= 8; off <<= 1) tot += __shfl_xor(tot, off, 32);

    const float T   = tot * (1.0f / 64.0f);     // sum_b mean-over-pixels
    const float inv = 1.0f / (T + PE_EPS);

    float ent = 0.0f;
    #pragma unroll
    for (int g = 0; g < 16; ++g) {
        const float p = (s[g] * (1.0f / 64.0f)) * inv;
        ent -= p * __logf(p + PE_EPS);
    }
    #pragma unroll
    for (int off = 1; off <= 8; off <<= 1) ent += __shfl_xor(ent, off, 32);

    if (lane == 0) out[P] = ent;
}

extern "C" void kernel_launch(void* const* d_in, const int* in_sizes, int n_in,
                              void* d_out, int out_size, void* d_ws, size_t ws_size,
                              hipStream_t stream) {
    (void)in_sizes; (void)n_in; (void)d_ws; (void)ws_size; (void)out_size;
    // setup_inputs() order: d_in[0] = coords (unused), d_in[1] = images (f32).
    const float* images = (const float*)d_in[1];
    float*       out    = (float*)d_out;       // 2048 f32 entropies (2,4,1,16,16)
    // 2048 patches, 1 wave per patch, 8 waves per 256-thread block -> 256 blocks.
    PatchInfoGainLoss_82085414961155_kernel<<<256, 256, 0, stream>>>(images, out);
}